// PETMLIPWrapper_84310208021026
// MI455X (gfx1250) — compile-verified
//
#include <hip/hip_runtime.h>
#include <hip/hip_bf16.h>

#define N_ATOMS_C 100000
#define K_NBR_C   64
#define N_STRUCT_C 32
#define HID_C     128

typedef __attribute__((ext_vector_type(2))) float v2f;
typedef __attribute__((ext_vector_type(8))) float v8f;

// Branch-free tanh: CDNA5 V_TANH_F32 transcendental (confirmed lowering),
// fallback = 1 - 2/(exp(2x)+1) via V_EXP_F32 + V_RCP_F32 (also branch-free).
__device__ __forceinline__ float fast_tanh(float v) {
#if __has_builtin(__builtin_amdgcn_tanhf)
    return __builtin_amdgcn_tanhf(v);
#else
    const float e = __builtin_amdgcn_exp2f(v * 2.8853900817779268f);
    return 1.0f - 2.0f * __builtin_amdgcn_rcpf(e + 1.0f);
#endif
}

// Scheduling fence (MI-scheduler level).
__device__ __forceinline__ void sched_fence() {
#if __has_builtin(__builtin_amdgcn_sched_barrier)
    __builtin_amdgcn_sched_barrier(0);
#endif
}

// Hard lifetime pin: forces every accumulator to be materialized HERE, so the
// per-tile tanh values die inside their own tile section (IR passes cannot
// sink the accumulation FMAs past this, unlike sched_barrier which only
// constrains the machine scheduler).
#define PIN_ACC()                                                              \
    asm volatile(""                                                            \
        : "+v"(e_acc),                                                         \
          "+v"(gx0[0]), "+v"(gx0[1]), "+v"(gx0[2]), "+v"(gx0[3]),              \
          "+v"(gx0[4]), "+v"(gx0[5]), "+v"(gx0[6]), "+v"(gx0[7]),              \
          "+v"(gx1[0]), "+v"(gx1[1]), "+v"(gx1[2]), "+v"(gx1[3]),              \
          "+v"(gx1[4]), "+v"(gx1[5]), "+v"(gx1[6]), "+v"(gx1[7]),              \
          "+v"(gx2[0]), "+v"(gx2[1]), "+v"(gx2[2]), "+v"(gx2[3]),              \
          "+v"(gx2[4]), "+v"(gx2[5]), "+v"(gx2[6]), "+v"(gx2[7]))

struct f3 { float a, b, c; };   // 12-byte POD for b96-style stores/loads

// ---------------------------------------------------------------------------
// Kernel 0: zero the preds slot of d_out (harness poisons it with 0xAA).
// ---------------------------------------------------------------------------
__global__ void pet_init_preds(float* __restrict__ preds) {
    if (threadIdx.x < N_STRUCT_C) preds[threadIdx.x] = 0.0f;
}

// ---------------------------------------------------------------------------
// Kernel 1: fused forward + input-gradient per edge.
// One wave per atom (64 edges, 4 edge-groups of 16).
// z = x @ W1 (3->4 padded) via V_WMMA_F32_16X16X4_F32; pointwise tanh/grad on
// the D layout (lane = hidden col, VGPR r = edge row) where w2/W1 multipliers
// are per-lane scalars. Mask folds into the per-element multiplier mw BEFORE
// accumulation, so stored grads are pre-masked and the energy / first-force
// sums accumulate as per-lane partials reduced ONCE per atom.
// __launch_bounds__(256,4): 256-VGPR budget (addressable window, no
// s_set_vgpr_msb churn) + 4 waves/SIMD; PIN_ACC bounds liveness to fit it.
// ---------------------------------------------------------------------------
__global__ void __launch_bounds__(256, 4)
pet_fwd_bwd(const float* __restrict__ x,            // [N,K,3]
            const unsigned char* __restrict__ mask, // [N,K]
            const int* __restrict__ batch_idx,      // [N]
            const float* __restrict__ W1,           // [3,HID]
            const float* __restrict__ w2,           // [HID]
            float* __restrict__ grads,              // ws [N,K,3]
            float* __restrict__ firstv,             // ws [N,3]
            float* __restrict__ preds)              // d_out[0:32]
{
    __shared__ float sbins[N_STRUCT_C];
    const int tid = threadIdx.x;
    if (tid < N_STRUCT_C) sbins[tid] = 0.0f;
    __syncthreads();

    const int wave = tid >> 5;
    const int lane = tid & 31;
    const int n = blockIdx.x * 8 + wave;

    if (n < N_ATOMS_C) {                       // wave-uniform: EXEC stays all-1s
        const int j  = lane & 15;
        const bool hi = lane >= 16;

        // Stationary operands: 8 B-tiles of W1 (4x16 each), per-lane w2/W1 cols.
        v2f  b[8];
        float w2l[8], w1l0[8], w1l1[8], w1l2[8];
#pragma unroll
        for (int t = 0; t < 8; ++t) {
            const int col = t * 16 + j;
            const float r0 = W1[0 * HID_C + col];
            const float r1 = W1[1 * HID_C + col];
            const float r2 = W1[2 * HID_C + col];
            b[t].x = hi ? r2 : r0;             // K=2 row / K=0 row
            b[t].y = hi ? 0.0f : r1;           // K=3 pad / K=1 row
            w2l[t] = w2[col];
            w1l0[t] = r0; w1l1[t] = r1; w1l2[t] = r2;
        }

        // Per-lane partial accumulators (valid on every lane; reduced once).
        float e_acc = 0.0f, f0 = 0.0f, f1 = 0.0f, f2 = 0.0f;

        for (int eg = 0; eg < 4; ++eg) {
            // A operand: one uniform b64 load per lane; cndmask assembles
            // {x0,x1} (lo lanes, K=0,1) / {x2,0} (hi lanes, K=2,pad).
            const int ka = eg * 16 + j;
            const float* px = x + ((size_t)n * K_NBR_C + ka) * 3 + (hi ? 1 : 0);
            v2f v = *(const v2f*)px;
            v2f a;
            a.x = hi ? v.y : v.x;
            a.y = hi ? 0.0f : v.y;

            // Mask bytes for this half's 8 edge rows (bytes are exactly 0/1):
            // msel = 1 - byte, pure arithmetic (no wide compares/predicates).
            const uint2 mu = *(const uint2*)(mask + (size_t)n * K_NBR_C +
                                             eg * 16 + (hi ? 8 : 0));
            float msel[8];
#pragma unroll
            for (int r = 0; r < 4; ++r) {
                msel[r]     = 1.0f - (float)((mu.x >> (8 * r)) & 0xffu);
                msel[r + 4] = 1.0f - (float)((mu.y >> (8 * r)) & 0xffu);
            }

            float gx0[8], gx1[8], gx2[8];
#pragma unroll
            for (int r = 0; r < 8; ++r) { gx0[r] = 0.f; gx1[r] = 0.f; gx2[r] = 0.f; }

#pragma unroll
            for (int t = 0; t < 8; ++t) {
                v8f z = {};
                z = __builtin_amdgcn_wmma_f32_16x16x4_f32(
                        false, a, false, b[t], (short)0, z, false, false);
#pragma unroll
                for (int r = 0; r < 8; ++r) {
                    const float h  = fast_tanh(z[r]);
                    const float mw = msel[r] * w2l[t];       // masked w2
                    e_acc = fmaf(h, mw, e_acc);              // masked energy
                    const float gd = (1.0f - h * h) * mw;    // masked grad elem
                    gx0[r] += gd * w1l0[t];
                    gx1[r] += gd * w1l1[t];
                    gx2[r] += gd * w1l2[t];
                }
                PIN_ACC();       // materialize accumulators; h values die here
                sched_fence();
            }

            // First-force per-lane partials (pre-reduction values).
#pragma unroll
            for (int r = 0; r < 8; ++r) { f0 += gx0[r]; f1 += gx1[r]; f2 += gx2[r]; }

            // Butterfly reduce gx per edge across the 16 lanes of each half.
#pragma unroll
            for (int off = 1; off < 16; off <<= 1) {
#pragma unroll
                for (int r = 0; r < 8; ++r) {
                    gx0[r] += __shfl_xor(gx0[r], off, 32);
                    gx1[r] += __shfl_xor(gx1[r], off, 32);
                    gx2[r] += __shfl_xor(gx2[r], off, 32);
                }
            }

            if (j == 0) {                      // lane 0 -> edges 0..7, lane 16 -> 8..15
#pragma unroll
                for (int r = 0; r < 8; ++r) {
                    const int k = eg * 16 + r + (hi ? 8 : 0);
                    f3 g; g.a = gx0[r]; g.b = gx1[r]; g.c = gx2[r];
                    *(f3*)(grads + ((size_t)n * K_NBR_C + k) * 3) = g;
                }
            }
            sched_fence();       // do not pipeline across edge-groups
        }

        // One full-wave reduction per atom (both halves hold distinct edges /
        // distinct hidden columns, so summing all 32 lanes is exact).
#pragma unroll
        for (int off = 1; off < 32; off <<= 1) {
            e_acc += __shfl_xor(e_acc, off, 32);
            f0    += __shfl_xor(f0,    off, 32);
            f1    += __shfl_xor(f1,    off, 32);
            f2    += __shfl_xor(f2,    off, 32);
        }

        if (lane == 0) {
            const size_t fb = (size_t)n * 3;
            firstv[fb + 0] = f0;
            firstv[fb + 1] = f1;
            firstv[fb + 2] = f2;
            atomicAdd(&sbins[batch_idx[n]], e_acc);
        }
    }

    __syncthreads();
    if (tid < N_STRUCT_C) atomicAdd(&preds[tid], sbins[tid]);
}

// ---------------------------------------------------------------------------
// Kernel 2: reverse-edge gather + forces. One wave per atom; lane l handles
// edges l and l+32. second = sum_k !mask * grads[nbr,pos,:]; forces = first-second.
// Random gather hits the 192MB L2 (grads buffer is 77MB, fully resident).
// ---------------------------------------------------------------------------
__global__ void __launch_bounds__(256, 4)
pet_gather_forces(const int* __restrict__ nbr_index,  // [K,N]
                  const int* __restrict__ nbr_pos,    // [N,K]
                  const unsigned char* __restrict__ mask, // [N,K]
                  const float* __restrict__ grads,    // ws [N,K,3]
                  const float* __restrict__ firstv,   // ws [N,3]
                  float* __restrict__ forces)         // d_out+32
{
    const int wave = threadIdx.x >> 5;
    const int lane = threadIdx.x & 31;
    const int n = blockIdx.x * 8 + wave;
    if (n >= N_ATOMS_C) return;

    float s0 = 0.0f, s1 = 0.0f, s2 = 0.0f;
#pragma unroll
    for (int half = 0; half < 2; ++half) {
        const int k = lane + half * 32;
        if (mask[(size_t)n * K_NBR_C + k] == 0) {
            const int nb = nbr_index[(size_t)k * N_ATOMS_C + n];
            const int p  = nbr_pos[(size_t)n * K_NBR_C + k];
            const f3 g = *(const f3*)(grads + ((size_t)nb * K_NBR_C + p) * 3);
            s0 += g.a; s1 += g.b; s2 += g.c;
        }
    }
#pragma unroll
    for (int off = 1; off < 32; off <<= 1) {
        s0 += __shfl_xor(s0, off, 32);
        s1 += __shfl_xor(s1, off, 32);
        s2 += __shfl_xor(s2, off, 32);
    }
    if (lane == 0) {
        const size_t fb = (size_t)n * 3;
        forces[fb + 0] = firstv[fb + 0] - s0;
        forces[fb + 1] = firstv[fb + 1] - s1;
        forces[fb + 2] = firstv[fb + 2] - s2;
    }
}

// ---------------------------------------------------------------------------
// Launcher. Inputs (setup_inputs order): x, neighbors_index, neighbors_pos,
// mask, batch_idx, W1, w2. Output: [preds(32) | forces(N*3)] float32.
// Workspace: grads N*K*3 floats (76.8MB) + first N*3 floats (1.2MB).
// ---------------------------------------------------------------------------
extern "C" void kernel_launch(void* const* d_in, const int* in_sizes, int n_in,
                              void* d_out, int out_size, void* d_ws, size_t ws_size,
                              hipStream_t stream) {
    (void)in_sizes; (void)n_in; (void)out_size; (void)ws_size;

    const float*         x         = (const float*)d_in[0];
    const int*           nbr_index = (const int*)d_in[1];
    const int*           nbr_pos   = (const int*)d_in[2];
    const unsigned char* mask      = (const unsigned char*)d_in[3];
    const int*           batch_idx = (const int*)d_in[4];
    const float*         W1        = (const float*)d_in[5];
    const float*         w2        = (const float*)d_in[6];

    float* preds  = (float*)d_out;
    float* forces = preds + N_STRUCT_C;

    float* grads  = (float*)d_ws;                                   // [N,K,3]
    float* firstv = grads + (size_t)N_ATOMS_C * K_NBR_C * 3;        // [N,3]

    pet_init_preds<<<1, 32, 0, stream>>>(preds);

    const int wavesPerBlock = 8;                 // 256 threads = 8 wave32s
    const int blocks = (N_ATOMS_C + wavesPerBlock - 1) / wavesPerBlock;

    pet_fwd_bwd<<<blocks, 256, 0, stream>>>(x, mask, batch_idx, W1, w2,
                                            grads, firstv, preds);
    pet_gather_forces<<<blocks, 256, 0, stream>>>(nbr_index, nbr_pos, mask,
                                                  grads, firstv, forces);
}